// HGTAttention_850403524851
// MI455X (gfx1250) — compile-verified
//
#include <hip/hip_runtime.h>
#include <hip/hip_bf16.h>

#define HIDDEN   128
#define HEADS    8
#define HEAD_DIM 16
#define INV_SCALE 0.25f   /* 1/sqrt(16) */

typedef float v2f __attribute__((ext_vector_type(2)));
typedef float v8f __attribute__((ext_vector_type(8)));

// ---------------------------------------------------------------------------
// Zero-fill helper (graph-capture safe; we own all init)
// ---------------------------------------------------------------------------
__global__ void hgt_zero_f32(float* __restrict__ p, int n) {
    int i = blockIdx.x * blockDim.x + threadIdx.x;
    if (i < n) p[i] = 0.0f;
}

// ---------------------------------------------------------------------------
// Fold the (scalar-typed) relation matrices into the K / V projection
// weights, and transpose q_w into K-major layout.
//   Weff_k[j, h*16+o] = sum_i k_w[st][(h*16+i)*128 + j] * Ratt[et][h,i,o]
//   Weff_v analogous with Rmsg;  Wq_eff[j,c] = q_w[dt][c,j]
// ---------------------------------------------------------------------------
__global__ __launch_bounds__(256) void hgt_fuse_weights(
    const int* __restrict__ st_p, const int* __restrict__ et_p, const int* __restrict__ dt_p,
    const float* __restrict__ k_w, const float* __restrict__ k_b,
    const float* __restrict__ q_w, const float* __restrict__ q_b,
    const float* __restrict__ v_w, const float* __restrict__ v_b,
    const float* __restrict__ rel_att, const float* __restrict__ rel_msg,
    float* __restrict__ Wk, float* __restrict__ bk,
    float* __restrict__ Wv, float* __restrict__ bv,
    float* __restrict__ Wq, float* __restrict__ bq)
{
    const int st = st_p[0], et = et_p[0], dt = dt_p[0];
    const float* kw = k_w + (size_t)st * HIDDEN * HIDDEN;
    const float* vw = v_w + (size_t)st * HIDDEN * HIDDEN;
    const float* qw = q_w + (size_t)dt * HIDDEN * HIDDEN;
    const float* Ra = rel_att + (size_t)et * HEADS * HEAD_DIM * HEAD_DIM;
    const float* Rm = rel_msg + (size_t)et * HEADS * HEAD_DIM * HEAD_DIM;

    for (int p = threadIdx.x; p < HIDDEN * HIDDEN; p += blockDim.x) {
        int j = p >> 7, c = p & 127;
        int h = c >> 4, o = c & 15;
        float sk = 0.0f, sv = 0.0f;
        #pragma unroll
        for (int i = 0; i < HEAD_DIM; ++i) {
            float ra = Ra[(h * 16 + i) * 16 + o];
            float rm = Rm[(h * 16 + i) * 16 + o];
            sk += kw[(h * 16 + i) * 128 + j] * ra;
            sv += vw[(h * 16 + i) * 128 + j] * rm;
        }
        Wk[p] = sk;
        Wv[p] = sv;
        Wq[p] = qw[c * 128 + j];          // transpose to K-major
    }
    for (int c = threadIdx.x; c < HIDDEN; c += blockDim.x) {
        int h = c >> 4, o = c & 15;
        float sk = 0.0f, sv = 0.0f;
        #pragma unroll
        for (int i = 0; i < HEAD_DIM; ++i) {
            sk += k_b[st * 128 + h * 16 + i] * Ra[(h * 16 + i) * 16 + o];
            sv += v_b[st * 128 + h * 16 + i] * Rm[(h * 16 + i) * 16 + o];
        }
        bk[c] = sk;
        bv[c] = sv;
        bq[c] = q_b[dt * 128 + c];
    }
}

// ---------------------------------------------------------------------------
// C[N,128] = A[N,128] @ W[128,128] + bias   (W is K-major: W[k*128 + n])
// fp32 WMMA path: V_WMMA_F32_16X16X4_F32, one wave per 16x16 output tile,
// 8 waves/block cover all 128 columns of one 16-row tile. A tile staged in
// LDS with stride 132 floats (conflict-free column reads: bank step of 4).
// Full-tile fast path (uniform branch) keeps loads/stores unconditional so
// the epilogue is a clause of 8 global_store_b32 with immediate offsets.
// ---------------------------------------------------------------------------
__global__ __launch_bounds__(256) void hgt_gemm128_wmma(
    const float* __restrict__ A, const float* __restrict__ W,
    const float* __restrict__ bias, float* __restrict__ C, int N)
{
    __shared__ float As[16 * 132];
    const int row0 = blockIdx.x * 16;
    const int tid  = threadIdx.x;
    const bool full = (row0 + 16 <= N);      // uniform across block

    // cooperative coalesced load of the 16x128 A tile (2048 elem / 256 thr)
    if (full) {
        #pragma unroll
        for (int s = 0; s < 8; ++s) {
            int idx = tid + s * 256;
            int r = idx >> 7, c = idx & 127;
            As[r * 132 + c] = A[(size_t)(row0 + r) * 128 + c];
        }
    } else {
        #pragma unroll
        for (int s = 0; s < 8; ++s) {
            int idx = tid + s * 256;
            int r = idx >> 7, c = idx & 127;
            int row = row0 + r;
            As[r * 132 + c] = (row < N) ? A[(size_t)row * 128 + c] : 0.0f;
        }
    }
    __syncthreads();

    const int wave = tid >> 5;        // 0..7 -> column tile
    const int lane = tid & 31;
    const int lr   = lane & 15;       // 0..15
    const int lh   = lane >> 4;       // 0 or 1
    const int col  = wave * 16 + lr;

    v8f acc;
    {
        float b0 = bias[col];
        #pragma unroll
        for (int r = 0; r < 8; ++r) acc[r] = b0;   // D[r+8*lh][col] gets bias[col]
    }

    const float* __restrict__ wp = W + col;        // per-lane B column base
    const float* __restrict__ ap = As + lr * 132 + 2 * lh;

    // K = 128 in 32 steps of 4
    #pragma unroll 8
    for (int k = 0; k < 32; ++k) {
        const int kk = k * 4;
        v2f a, b;
        // A 16x4 frag: lane lr holds row lr; vgpr0/1 = K = 2*lh / 2*lh+1
        a.x = ap[kk];
        a.y = ap[kk + 1];
        // B 4x16 frag: vgpr0/1 = K = 2*lh / 2*lh+1, N = lane%16
        b.x = wp[(kk + 2 * lh) * 128];
        b.y = wp[(kk + 2 * lh + 1) * 128];
        acc = __builtin_amdgcn_wmma_f32_16x16x4_f32(false, a, false, b,
                                                    (short)0, acc, false, false);
    }

    // C layout: vgpr r -> M = r + 8*(lane/16)
    float* __restrict__ cp = C + (size_t)(row0 + 8 * lh) * 128 + col;
    if (full) {
        #pragma unroll
        for (int r = 0; r < 8; ++r) cp[r * 128] = acc[r];
    } else {
        #pragma unroll
        for (int r = 0; r < 8; ++r)
            if (row0 + 8 * lh + r < N) cp[r * 128] = acc[r];
    }
}

// ---------------------------------------------------------------------------
// Per-(edge, head) attention score + segmented atomic max (as uint; scores
// clamped to >=0 first, matching reference's max(0, seg_max) semantics).
// ---------------------------------------------------------------------------
__global__ __launch_bounds__(256) void hgt_edge_att(
    const float* __restrict__ K2, const float* __restrict__ Q,
    const int* __restrict__ ei, const float* __restrict__ pri,
    const int* __restrict__ et_p,
    float* __restrict__ att, unsigned int* __restrict__ seg_max, int n_edges)
{
    int t = blockIdx.x * blockDim.x + threadIdx.x;
    if (t >= n_edges * HEADS) return;
    int e = t >> 3, h = t & 7;
    int src = ei[e], dst = ei[n_edges + e];

    const float4* pk = (const float4*)(K2 + (size_t)src * 128 + h * 16);
    const float4* pq = (const float4*)(Q  + (size_t)dst * 128 + h * 16);
    float s = 0.0f;
    #pragma unroll
    for (int i = 0; i < 4; ++i) {
        float4 a = pk[i], b = pq[i];
        s += a.x * b.x + a.y * b.y + a.z * b.z + a.w * b.w;
    }
    float a = s * INV_SCALE + pri[et_p[0] * HEADS + h];
    att[t] = a;
    // nonnegative floats order identically to their uint bit patterns
    atomicMax(&seg_max[(size_t)dst * HEADS + h], __float_as_uint(fmaxf(a, 0.0f)));
}

// ---------------------------------------------------------------------------
// att -> exp(att - seg_max[dst]); atomic segmented sum.
// ---------------------------------------------------------------------------
__global__ __launch_bounds__(256) void hgt_edge_exp(
    const int* __restrict__ ei, float* __restrict__ att,
    const unsigned int* __restrict__ seg_max, float* __restrict__ seg_sum,
    int n_edges)
{
    int t = blockIdx.x * blockDim.x + threadIdx.x;
    if (t >= n_edges * HEADS) return;
    int e = t >> 3, h = t & 7;
    int dst = ei[n_edges + e];
    float m = __uint_as_float(seg_max[(size_t)dst * HEADS + h]);
    float v = expf(att[t] - m);
    att[t] = v;
    atomicAdd(&seg_sum[(size_t)dst * HEADS + h], v);
}

// ---------------------------------------------------------------------------
// out[dst, c0..c0+3] += (att_exp / max(seg_sum,1e-8)) * V2[src, c0..c0+3].
// One thread per (edge, 4-channel quad): float4 V2 load, 4 f32 atomics with
// immediate offsets; att/seg_sum fetched once per quad (not per channel).
// ---------------------------------------------------------------------------
__global__ __launch_bounds__(256) void hgt_edge_agg(
    const int* __restrict__ ei, const float* __restrict__ att,
    const float* __restrict__ seg_sum, const float* __restrict__ V2,
    float* __restrict__ out, int n_edges)
{
    unsigned int t = blockIdx.x * blockDim.x + threadIdx.x;
    if (t >= (unsigned int)n_edges * 32u) return;
    int e = t >> 5, q = t & 31;        // q: which float4 of the 128-wide row
    int h = q >> 2;                    // 4 quads per head
    int src = ei[e], dst = ei[n_edges + e];
    float w = att[(size_t)e * HEADS + h] /
              fmaxf(seg_sum[(size_t)dst * HEADS + h], 1e-8f);
    float4 v = ((const float4*)(V2 + (size_t)src * 128))[q];
    float* o = out + (size_t)dst * 128 + q * 4;
    atomicAdd(o + 0, w * v.x);
    atomicAdd(o + 1, w * v.y);
    atomicAdd(o + 2, w * v.z);
    atomicAdd(o + 3, w * v.w);
}

// ---------------------------------------------------------------------------
extern "C" void kernel_launch(void* const* d_in, const int* in_sizes, int n_in,
                              void* d_out, int out_size, void* d_ws, size_t ws_size,
                              hipStream_t stream) {
    const float* x_src   = (const float*)d_in[0];
    const float* x_dst   = (const float*)d_in[1];
    const int*   ei      = (const int*)d_in[2];
    const int*   st_p    = (const int*)d_in[3];
    const int*   et_p    = (const int*)d_in[4];
    const int*   dt_p    = (const int*)d_in[5];
    const float* k_w     = (const float*)d_in[6];
    const float* k_b     = (const float*)d_in[7];
    const float* q_w     = (const float*)d_in[8];
    const float* q_b     = (const float*)d_in[9];
    const float* v_w     = (const float*)d_in[10];
    const float* v_b     = (const float*)d_in[11];
    const float* rel_att = (const float*)d_in[12];
    const float* rel_msg = (const float*)d_in[13];
    const float* rel_pri = (const float*)d_in[14];

    const int n_src   = in_sizes[0] / HIDDEN;
    const int n_dst   = in_sizes[1] / HIDDEN;
    const int n_edges = in_sizes[2] / 2;

    // ---- workspace carve-up (offsets up to att are multiples of 128 floats) ----
    float* ws = (float*)d_ws;
    float* Wk = ws;                    // 16384
    float* Wv = Wk + 16384;            // 16384
    float* Wq = Wv + 16384;            // 16384
    float* bk = Wq + 16384;            // 128
    float* bv = bk + 128;              // 128
    float* bq = bv + 128;              // 128
    float* K2 = bq + 128;              // n_src*128
    float* V2 = K2 + (size_t)n_src * 128;
    float* Qp = V2 + (size_t)n_src * 128;
    float* att = Qp + (size_t)n_dst * 128;
    unsigned int* seg_max = (unsigned int*)(att + (size_t)n_edges * HEADS);
    float* seg_sum = (float*)(seg_max + (size_t)n_dst * HEADS);
    float* out = (float*)d_out;

    // ---- init (deterministic every call) ----
    {
        int n1 = n_dst * HEADS;
        hgt_zero_f32<<<(n1 + 255) / 256, 256, 0, stream>>>((float*)seg_max, n1);
        hgt_zero_f32<<<(n1 + 255) / 256, 256, 0, stream>>>(seg_sum, n1);
        hgt_zero_f32<<<(out_size + 255) / 256, 256, 0, stream>>>(out, out_size);
    }

    // ---- fold relation matrices into projection weights ----
    hgt_fuse_weights<<<1, 256, 0, stream>>>(st_p, et_p, dt_p,
                                            k_w, k_b, q_w, q_b, v_w, v_b,
                                            rel_att, rel_msg,
                                            Wk, bk, Wv, bv, Wq, bq);

    // ---- WMMA GEMMs: K2/V2 over src nodes, Q over dst nodes ----
    hgt_gemm128_wmma<<<(n_src + 15) / 16, 256, 0, stream>>>(x_src, Wk, bk, K2, n_src);
    hgt_gemm128_wmma<<<(n_src + 15) / 16, 256, 0, stream>>>(x_src, Wv, bv, V2, n_src);
    hgt_gemm128_wmma<<<(n_dst + 15) / 16, 256, 0, stream>>>(x_dst, Wq, bq, Qp, n_dst);

    // ---- edge phase: scores, segmented softmax, weighted scatter-add ----
    {
        int nt = n_edges * HEADS;
        hgt_edge_att<<<(nt + 255) / 256, 256, 0, stream>>>(K2, Qp, ei, rel_pri, et_p,
                                                           att, seg_max, n_edges);
        hgt_edge_exp<<<(nt + 255) / 256, 256, 0, stream>>>(ei, att, seg_max, seg_sum,
                                                           n_edges);
        unsigned int na = (unsigned int)n_edges * 32u;
        hgt_edge_agg<<<(na + 255) / 256, 256, 0, stream>>>(ei, att, seg_sum, V2,
                                                           out, n_edges);
    }
}